// NeuralSLDS_54692113547703
// MI455X (gfx1250) — compile-verified
//
#include <hip/hip_runtime.h>

typedef float v2f __attribute__((ext_vector_type(2)));
typedef float v4f __attribute__((ext_vector_type(4)));
typedef float v8f __attribute__((ext_vector_type(8)));

#define Bb   512
#define Tt   2048
#define Kk   8
#define Dd   32
#define DT_  0.05f
#define SQDT 0.22360679774997896f   /* sqrt(0.05) */

#define NJOB 128   /* 64 outer-product K-chunks x 2 drift N-tiles (LDS-resident) */

#define WMMA_F32(A, Bv, C) \
  __builtin_amdgcn_wmma_f32_16x16x4_f32(false, (A), false, (Bv), (short)0, (C), false, false)

__global__ __launch_bounds__(32)
void slds_scan_kernel(const float* __restrict__ z0,
                      const float* __restrict__ s_probs,
                      const float* __restrict__ noise,
                      const float* __restrict__ A_s,
                      const float* __restrict__ b_s,
                      const float* __restrict__ Q_chol,
                      float* __restrict__ ys)
{
  // Pre-swizzled A_s (time-invariant), already in WMMA B-operand layout:
  // job jb=(2c+nt), lane l -> v2f of rows kd=(4c + 2*half + {0,1}), col nt*16+(l&15).
  __shared__ v2f   swz[NJOB * 32];    // 32768 B
  __shared__ float ldsz[16 * 32];     // z transpose buffer, 2048 B

  const int lane = threadIdx.x & 31;
  const int lid  = lane & 15;
  const int half = lane >> 4;
  const int tile = blockIdx.x * 16;

  // ---------------- stage swizzled A_s into LDS (once) ----------------
  for (int jb = 0; jb < NJOB; ++jb) {
    const int c    = jb >> 1;
    const int nt   = jb & 1;
    const int ncol = nt * 16 + lid;          // output column i
    const int kd0  = 4 * c + 2 * half;       // kd = k*32 + j
    v2f val;
    {
      const int k = kd0 >> 5, j = kd0 & 31;
      val.x = A_s[(k * Dd + ncol) * Dd + j];
      val.y = A_s[(k * Dd + ncol) * Dd + (j + 1)];
    }
    swz[jb * 32 + lane] = val;
  }
  __syncthreads();

  // ---- time-invariant bias/diffusion B-operands kept in registers ----
  // bias chunks c=64,65: rows kd=256+k hold b_s (drift cols) / Q_chol (diff cols).
  v2f rBd[4], rBq[4];   // [c64nt0, c64nt1, c65nt0, c65nt1]
  {
    const int k0 = 2 * half;        // chunk 64 -> k = 2*half + s
    const int k1 = 4 + 2 * half;    // chunk 65 -> k = 4 + 2*half + s
    rBd[0].x = b_s[k0 * Dd + lid];            rBd[0].y = b_s[(k0 + 1) * Dd + lid];
    rBd[1].x = b_s[k0 * Dd + 16 + lid];       rBd[1].y = b_s[(k0 + 1) * Dd + 16 + lid];
    rBd[2].x = b_s[k1 * Dd + lid];            rBd[2].y = b_s[(k1 + 1) * Dd + lid];
    rBd[3].x = b_s[k1 * Dd + 16 + lid];       rBd[3].y = b_s[(k1 + 1) * Dd + 16 + lid];
    rBq[0].x = Q_chol[k0 * Dd + lid];         rBq[0].y = Q_chol[(k0 + 1) * Dd + lid];
    rBq[1].x = Q_chol[k0 * Dd + 16 + lid];    rBq[1].y = Q_chol[(k0 + 1) * Dd + 16 + lid];
    rBq[2].x = Q_chol[k1 * Dd + lid];         rBq[2].y = Q_chol[(k1 + 1) * Dd + lid];
    rBq[3].x = Q_chol[k1 * Dd + 16 + lid];    rBq[3].y = Q_chol[(k1 + 1) * Dd + 16 + lid];
  }

  // ---------------- initial state ----------------
  float zc0[8], zc1[8];               // z in C layout (lane=col, reg=row, half=+8 rows)
  #pragma unroll
  for (int r = 0; r < 8; ++r) {
    const int bm = tile + r + half * 8;
    zc0[r] = z0[bm * Dd + lid];
    zc1[r] = z0[bm * Dd + 16 + lid];
  }
  float zs0[8], zs1[8];               // z in A-operand form (lane owns batch lid)
  {
    const int bm = tile + lid;
    #pragma unroll
    for (int tq = 0; tq < 8; ++tq) {
      v4f q = *(const v4f*)&z0[bm * Dd + 4 * tq];
      zs0[tq] = half ? q.z : q.x;
      zs1[tq] = half ? q.w : q.y;
    }
  }

  // normalized weights for step 0, and streamed noise for step 0
  float wt[8], nz0[8], nz1[8];
  {
    const int bm = tile + lid;
    v4f wa = *(const v4f*)&s_probs[(size_t)bm * Kk];
    v4f wb = *(const v4f*)&s_probs[(size_t)bm * Kk + 4];
    const float wsum = ((wa.x + wa.y) + (wa.z + wa.w)) + ((wb.x + wb.y) + (wb.z + wb.w));
    const float inv  = 1.0f / wsum;
    wt[0] = wa.x * inv; wt[1] = wa.y * inv; wt[2] = wa.z * inv; wt[3] = wa.w * inv;
    wt[4] = wb.x * inv; wt[5] = wb.y * inv; wt[6] = wb.z * inv; wt[7] = wb.w * inv;
    #pragma unroll
    for (int r = 0; r < 8; ++r) {
      const int bmc = tile + r + half * 8;
      nz0[r] = SQDT * noise[(size_t)bmc * Dd + lid];
      nz1[r] = SQDT * noise[(size_t)bmc * Dd + 16 + lid];
    }
  }

  for (int t = 0; t < Tt; ++t) {
    // ---- issue next step's global loads first (hidden under WMMA chain) ----
    float wn[8]  = {0,0,0,0,0,0,0,0};
    float nn0[8] = {0,0,0,0,0,0,0,0};
    float nn1[8] = {0,0,0,0,0,0,0,0};
    if (t + 1 < Tt) {
      const int bm = tile + lid;
      const size_t wbase = ((size_t)(t + 1) * Bb + bm) * Kk;
      v4f wa = *(const v4f*)&s_probs[wbase];
      v4f wb = *(const v4f*)&s_probs[wbase + 4];
      wn[0] = wa.x; wn[1] = wa.y; wn[2] = wa.z; wn[3] = wa.w;
      wn[4] = wb.x; wn[5] = wb.y; wn[6] = wb.z; wn[7] = wb.w;
      #pragma unroll
      for (int r = 0; r < 8; ++r) {
        const int bmc = tile + r + half * 8;
        const size_t nb = ((size_t)(t + 1) * Bb + bmc) * Dd;
        nn0[r] = SQDT * noise[nb + lid];
        nn1[r] = SQDT * noise[nb + 16 + lid];
      }
    }
    if (t + 4 < Tt) {   // deeper prefetch -> global_prefetch_b8
      __builtin_prefetch(&s_probs[((size_t)(t + 4) * Bb + tile + lid) * Kk], 0, 1);
      __builtin_prefetch(&noise[((size_t)(t + 4) * Bb + tile + lid) * Dd], 0, 1);
    }

    // ---- GEMM: [16 x 264] X @ [264 x 64] -> drift|diff in C layout ----
    // 4 independent drift accumulation chains (RAW distance 4 WMMAs).
    v8f acc0a = {0,0,0,0,0,0,0,0}, acc0b = {0,0,0,0,0,0,0,0};
    v8f acc1a = {0,0,0,0,0,0,0,0}, acc1b = {0,0,0,0,0,0,0,0};

    // software-pipelined B-operand fetch, depth 2
    v2f p0t0 = swz[0 * 32 + lane], p0t1 = swz[1 * 32 + lane];
    v2f p1t0 = swz[2 * 32 + lane], p1t1 = swz[3 * 32 + lane];

    #pragma unroll
    for (int c = 0; c < 64; ++c) {
      const v2f c0 = p0t0, c1 = p0t1;
      p0t0 = p1t0; p0t1 = p1t1;
      if (c + 2 < 64) {
        p1t0 = swz[(2 * (c + 2) + 0) * 32 + lane];
        p1t1 = swz[(2 * (c + 2) + 1) * 32 + lane];
      }
      const int kc = c >> 3, tc = c & 7;
      v2f a;
      a.x = wt[kc] * zs0[tc];
      a.y = wt[kc] * zs1[tc];
      if (c & 1) {
        acc0b = WMMA_F32(a, c0, acc0b);
        acc1b = WMMA_F32(a, c1, acc1b);
      } else {
        acc0a = WMMA_F32(a, c0, acc0a);
        acc1a = WMMA_F32(a, c1, acc1a);
      }
    }

    // bias + diffusion chunks (B operands live in registers)
    v8f acc2 = {0,0,0,0,0,0,0,0};   // diff cols  0..15
    v8f acc3 = {0,0,0,0,0,0,0,0};   // diff cols 16..31
    {
      v2f a64; a64.x = half ? wt[2] : wt[0]; a64.y = half ? wt[3] : wt[1];
      v2f a65; a65.x = half ? wt[6] : wt[4]; a65.y = half ? wt[7] : wt[5];
      acc0a = WMMA_F32(a64, rBd[0], acc0a);
      acc1a = WMMA_F32(a64, rBd[1], acc1a);
      acc0b = WMMA_F32(a65, rBd[2], acc0b);
      acc1b = WMMA_F32(a65, rBd[3], acc1b);
      acc2  = WMMA_F32(a64, rBq[0], acc2);
      acc3  = WMMA_F32(a64, rBq[1], acc3);
      acc2  = WMMA_F32(a65, rBq[2], acc2);
      acc3  = WMMA_F32(a65, rBq[3], acc3);
    }
    const v8f d0 = acc0a + acc0b;   // drift cols  0..15
    const v8f d1 = acc1a + acc1b;   // drift cols 16..31

    // ---- Euler-Maruyama update in C layout; emit ys[t]; stage transpose ----
    #pragma unroll
    for (int r = 0; r < 8; ++r) {
      const float zn0 = zc0[r] + DT_ * d0[r] + acc2[r] * nz0[r];
      const float zn1 = zc1[r] + DT_ * d1[r] + acc3[r] * nz1[r];
      zc0[r] = zn0; zc1[r] = zn1;
      const int bmc = tile + r + half * 8;
      const size_t ob = ((size_t)t * Bb + bmc) * Dd;
      ys[ob + lid]      = zn0;
      ys[ob + 16 + lid] = zn1;
      ldsz[(r + half * 8) * 32 + lid]      = zn0;
      ldsz[(r + half * 8) * 32 + 16 + lid] = zn1;
    }

    // overlap next step's weight normalization (precise div, ~25cyc) with the
    // LDS transpose round trip and the output stores
    {
      const float wsum = ((wn[0] + wn[1]) + (wn[2] + wn[3])) +
                         ((wn[4] + wn[5]) + (wn[6] + wn[7]));
      const float inv = 1.0f / wsum;
      #pragma unroll
      for (int k = 0; k < 8; ++k) wt[k] = wn[k] * inv;
    }

    __syncthreads();   // single wave: lowers to waitcnt; keeps LDS deps explicit
    #pragma unroll
    for (int tq = 0; tq < 8; ++tq) {
      v4f q = *(const v4f*)&ldsz[lid * 32 + 4 * tq];
      zs0[tq] = half ? q.z : q.x;
      zs1[tq] = half ? q.w : q.y;
    }
    __syncthreads();

    // ---- rotate streamed noise ----
    #pragma unroll
    for (int r = 0; r < 8; ++r) { nz0[r] = nn0[r]; nz1[r] = nn1[r]; }
  }
}

extern "C" void kernel_launch(void* const* d_in, const int* in_sizes, int n_in,
                              void* d_out, int out_size, void* d_ws, size_t ws_size,
                              hipStream_t stream) {
  (void)in_sizes; (void)n_in; (void)out_size; (void)d_ws; (void)ws_size;
  const float* z0      = (const float*)d_in[0];
  const float* s_probs = (const float*)d_in[1];
  const float* noise   = (const float*)d_in[2];
  const float* A_s     = (const float*)d_in[3];
  const float* b_s     = (const float*)d_in[4];
  const float* Q_chol  = (const float*)d_in[5];
  float* ys = (float*)d_out;

  dim3 grid(Bb / 16);   // 32 batch tiles
  dim3 block(32);       // one wave32 per tile
  slds_scan_kernel<<<grid, block, 0, stream>>>(z0, s_probs, noise, A_s, b_s, Q_chol, ys);
}